// LatentGMMRegistration_11862699672002
// MI455X (gfx1250) — compile-verified
//
#include <hip/hip_runtime.h>

// ---------------------------------------------------------------------------
// LatentGMMRegistration on MI455X (gfx1250)
//
// Roofline: gamma traffic = 2 * 32*16384*64*4 B = 268 MB, FLOPs ~0.5G
//   -> ~1.5 FLOP/byte -> purely HBM-bandwidth-bound (~12 us at 23.3 TB/s).
// Strategy: stream gamma once (non-temporal), fold the bnk,bnd->bkd einsum
// into fp32 WMMA (v_wmma_f32_16x16x4_f32): A = 16 clusters x 4 points of
// gamma, B = 4 points x [x,y,z,1,0...], f32 16x16 accumulator.
// This round: branchless B-fragment (clamped load + cndmask instead of
// exec-divergent loads) and dual accumulators to break the D->C chain.
// ---------------------------------------------------------------------------

typedef float v2f __attribute__((ext_vector_type(2)));
typedef float v8f __attribute__((ext_vector_type(8)));

constexpr int Bsz    = 32;
constexpr int Npts   = 16384;
constexpr int Kc     = 64;            // clusters
constexpr int KT     = 4;             // k-tiles of 16
constexpr int SLICES = 32;            // N-slices per (batch, k-tile)
constexpr int NS     = Npts / SLICES; // 512 points per wave

__global__ void zero_ws_kernel(float* p, int n) {
    int i = blockIdx.x * blockDim.x + threadIdx.x;
    if (i < n) p[i] = 0.0f;
}

// One wave32 per (b, ktile, slice). Accumulates S[b][k][0..2] = sum gamma*p,
// S[b][k][3] = sum gamma, via the fp32 WMMA path, atomically merged.
__global__ __launch_bounds__(128)
void moments_wmma(const float* __restrict__ gamma,   // [B][N][64]
                  const float* __restrict__ pts,     // [B][N][3]
                  float* __restrict__ S)             // [B][64][4]
{
    const int wid   = (blockIdx.x * blockDim.x + threadIdx.x) >> 5;
    const int lane  = threadIdx.x & 31;
    const int slice = wid % SLICES;
    const int ktile = (wid / SLICES) % KT;
    const int b     = wid / (SLICES * KT);

    const int g = lane >> 4;   // half-wave group (selects inner-K pair)
    const int c = lane & 15;   // A row (cluster) / B,C column

    const int    k0    = ktile * 16;
    const size_t gbase = (size_t)b * Npts * Kc + k0 + c;
    const size_t pbase = (size_t)b * Npts * 3;

    // Branchless column source: lanes c>=3 re-read coord 2 (same cacheline),
    // then a cndmask swaps in the [.,.,.,1,0...] padding.
    const int   cc    = (c < 2) ? c : 2;
    const float pad   = (c == 3) ? 1.0f : 0.0f;
    const bool  isxyz = (c < 3);

    v8f acc0 = {0.f, 0.f, 0.f, 0.f, 0.f, 0.f, 0.f, 0.f};
    v8f acc1 = {0.f, 0.f, 0.f, 0.f, 0.f, 0.f, 0.f, 0.f};

    const int n_begin = slice * NS;
    const int n_end   = n_begin + NS;

    #pragma unroll 2
    for (int n0 = n_begin; n0 < n_end; n0 += 8) {
        const int r0 = n0 + 2 * g;       // this half-wave's inner-K rows
        const int r1 = r0 + 4;

        // ---- group 0: points n0..n0+3 ----
        v2f a0;
        a0.x = __builtin_nontemporal_load(&gamma[gbase + (size_t)r0       * Kc]);
        a0.y = __builtin_nontemporal_load(&gamma[gbase + (size_t)(r0 + 1) * Kc]);
        const float p00 = pts[pbase + (size_t)r0       * 3 + cc];
        const float p01 = pts[pbase + (size_t)(r0 + 1) * 3 + cc];
        v2f b0;
        b0.x = isxyz ? p00 : pad;
        b0.y = isxyz ? p01 : pad;

        // ---- group 1: points n0+4..n0+7 ----
        v2f a1;
        a1.x = __builtin_nontemporal_load(&gamma[gbase + (size_t)r1       * Kc]);
        a1.y = __builtin_nontemporal_load(&gamma[gbase + (size_t)(r1 + 1) * Kc]);
        const float p10 = pts[pbase + (size_t)r1       * 3 + cc];
        const float p11 = pts[pbase + (size_t)(r1 + 1) * 3 + cc];
        v2f b1;
        b1.x = isxyz ? p10 : pad;
        b1.y = isxyz ? p11 : pad;

        acc0 = __builtin_amdgcn_wmma_f32_16x16x4_f32(
                   false, a0, false, b0, (short)0, acc0, false, false);
        acc1 = __builtin_amdgcn_wmma_f32_16x16x4_f32(
                   false, a1, false, b1, (short)0, acc1, false, false);
    }

    // C layout: VGPR r, lanes 0-15 -> M=r, lanes 16-31 -> M=r+8; col = c.
    if (c < 4) {
        #pragma unroll
        for (int r = 0; r < 8; ++r) {
            atomicAdd(S + ((size_t)b * Kc + k0 + r + 8 * g) * 4 + c,
                      acc0[r] + acc1[r]);
        }
    }
}

// One thread per batch: pi/mu, weighted centroids, covariance H, optimal
// proper rotation via Horn's quaternion (== SVD with diag(1,1,det) fix), t.
__global__ void solve_kernel(const float* __restrict__ Sx,
                             const float* __restrict__ Sy,
                             float* __restrict__ out)
{
    const int b = blockIdx.x * blockDim.x + threadIdx.x;
    if (b >= Bsz) return;
    const float EPS = 1e-8f;

    const float* sx = Sx + (size_t)b * Kc * 4;
    const float* sy = Sy + (size_t)b * Kc * 4;

    float M[3][3] = {{0}}, aS[3] = {0, 0, 0}, bS[3] = {0, 0, 0};
    float wsum = 0.f;
    for (int k = 0; k < Kc; ++k) {
        const float pix = sx[k * 4 + 3] * (1.0f / Npts);
        const float piy = sy[k * 4 + 3] * (1.0f / Npts);
        float mx[3], my[3];
        #pragma unroll
        for (int d = 0; d < 3; ++d) {
            mx[d] = sx[k * 4 + d] / (pix + EPS);
            my[d] = sy[k * 4 + d] / (piy + EPS);
        }
        const float w = pix * piy;
        wsum += w;
        #pragma unroll
        for (int i = 0; i < 3; ++i) {
            aS[i] += w * mx[i];
            bS[i] += w * my[i];
            #pragma unroll
            for (int j = 0; j < 3; ++j) M[i][j] += w * mx[i] * my[j];
        }
    }
    wsum += EPS;
    float cs[3], ct[3];
    #pragma unroll
    for (int i = 0; i < 3; ++i) { cs[i] = aS[i] / wsum; ct[i] = bS[i] / wsum; }

    // H = sum w (mx-cs)(my-ct)^T = M - aS bS^T / wsum
    float H[3][3];
    #pragma unroll
    for (int i = 0; i < 3; ++i)
        #pragma unroll
        for (int j = 0; j < 3; ++j)
            H[i][j] = M[i][j] - aS[i] * bS[j] / wsum;

    // Horn's 4x4 symmetric matrix; its top eigenvector is the quaternion of
    // the optimal proper rotation (matches SVD + reflection correction).
    const float Sxx = H[0][0], Sxy = H[0][1], Sxz = H[0][2];
    const float Syx = H[1][0], Syy = H[1][1], Syz = H[1][2];
    const float Szx = H[2][0], Szy = H[2][1], Szz = H[2][2];
    float Nq[4][4] = {
        {Sxx + Syy + Szz, Syz - Szy,        Szx - Sxz,        Sxy - Syx},
        {Syz - Szy,       Sxx - Syy - Szz,  Sxy + Syx,        Szx + Sxz},
        {Szx - Sxz,       Sxy + Syx,        Syy - Sxx - Szz,  Syz + Szy},
        {Sxy - Syx,       Szx + Sxz,        Syz + Szy,        Szz - Sxx - Syy}};
    float V[4][4] = {{1,0,0,0},{0,1,0,0},{0,0,1,0},{0,0,0,1}};

    for (int sweep = 0; sweep < 10; ++sweep) {
        #pragma unroll
        for (int p = 0; p < 3; ++p) {
            #pragma unroll
            for (int q = p + 1; q < 4; ++q) {
                const float apq = Nq[p][q];
                if (fabsf(apq) < 1e-12f) continue;
                const float theta = 0.5f * (Nq[q][q] - Nq[p][p]) / apq;
                const float tj = (theta >= 0.f ? 1.f : -1.f) /
                                 (fabsf(theta) + sqrtf(theta * theta + 1.f));
                const float cj = rsqrtf(tj * tj + 1.f);
                const float sj = tj * cj;
                #pragma unroll
                for (int i = 0; i < 4; ++i) {
                    const float nip = Nq[i][p], niq = Nq[i][q];
                    Nq[i][p] = cj * nip - sj * niq;
                    Nq[i][q] = sj * nip + cj * niq;
                }
                #pragma unroll
                for (int i = 0; i < 4; ++i) {
                    const float npi = Nq[p][i], nqi = Nq[q][i];
                    Nq[p][i] = cj * npi - sj * nqi;
                    Nq[q][i] = sj * npi + cj * nqi;
                }
                #pragma unroll
                for (int i = 0; i < 4; ++i) {
                    const float vip = V[i][p], viq = V[i][q];
                    V[i][p] = cj * vip - sj * viq;
                    V[i][q] = sj * vip + cj * viq;
                }
            }
        }
    }
    int   best = 0;
    float bl   = Nq[0][0];
    #pragma unroll
    for (int i = 1; i < 4; ++i)
        if (Nq[i][i] > bl) { bl = Nq[i][i]; best = i; }
    float qw = V[0][best], qx = V[1][best], qy = V[2][best], qz = V[3][best];
    const float nr = rsqrtf(qw * qw + qx * qx + qy * qy + qz * qz);
    qw *= nr; qx *= nr; qy *= nr; qz *= nr;

    const float R[3][3] = {
        {1 - 2*(qy*qy + qz*qz), 2*(qx*qy - qw*qz),     2*(qx*qz + qw*qy)},
        {2*(qx*qy + qw*qz),     1 - 2*(qx*qx + qz*qz), 2*(qy*qz - qw*qx)},
        {2*(qx*qz - qw*qy),     2*(qy*qz + qw*qx),     1 - 2*(qx*qx + qy*qy)}};

    float t3[3];
    #pragma unroll
    for (int i = 0; i < 3; ++i) {
        t3[i] = ct[i];
        #pragma unroll
        for (int j = 0; j < 3; ++j) t3[i] -= R[i][j] * cs[j];
    }
    #pragma unroll
    for (int i = 0; i < 3; ++i)
        #pragma unroll
        for (int j = 0; j < 3; ++j)
            out[(size_t)b * 9 + i * 3 + j] = R[i][j];
    #pragma unroll
    for (int i = 0; i < 3; ++i) out[(size_t)Bsz * 9 + b * 3 + i] = t3[i];
}

extern "C" void kernel_launch(void* const* d_in, const int* in_sizes, int n_in,
                              void* d_out, int out_size, void* d_ws, size_t ws_size,
                              hipStream_t stream) {
    const float* points_x = (const float*)d_in[0];
    const float* points_y = (const float*)d_in[1];
    const float* gamma_x  = (const float*)d_in[2];
    const float* gamma_y  = (const float*)d_in[3];
    float*       out      = (float*)d_out;

    float* Sx = (float*)d_ws;                 // [B][64][4]
    float* Sy = Sx + (size_t)Bsz * Kc * 4;    // [B][64][4]

    const int ztot = 2 * Bsz * Kc * 4;        // 16K floats of scratch
    zero_ws_kernel<<<(ztot + 255) / 256, 256, 0, stream>>>(Sx, ztot);

    const int waves  = Bsz * KT * SLICES;     // 4096 waves per tensor
    const int block  = 128;                   // 4 waves per block
    const int grid   = waves * 32 / block;

    moments_wmma<<<grid, block, 0, stream>>>(gamma_x, points_x, Sx);
    moments_wmma<<<grid, block, 0, stream>>>(gamma_y, points_y, Sy);
    solve_kernel<<<1, 32, 0, stream>>>(Sx, Sy, out);
}